// HypergraphNet_53618371723568
// MI455X (gfx1250) — compile-verified
//
#include <hip/hip_runtime.h>

namespace {
constexpr int kNodes = 100000;
constexpr int kEdges = 10000;
constexpr int kNNZ   = 1600000;
constexpr int kF     = 128;
}

typedef __attribute__((ext_vector_type(2))) float v2f;
typedef __attribute__((ext_vector_type(8))) float v8f;

// ---------------- scalar propagation kernels (bandwidth-bound path) ----------

__global__ void hg_zero(float* __restrict__ p, int n) {
  int i = blockIdx.x * blockDim.x + threadIdx.x;
  if (i < n) p[i] = 0.0f;
}

__global__ void hg_degree(const int* __restrict__ nidx, const int* __restrict__ eidx,
                          float* __restrict__ degD, float* __restrict__ degB) {
  int i = (blockIdx.x * blockDim.x + threadIdx.x) * 4;
  if (i >= kNNZ) return;
  int4 n4 = *reinterpret_cast<const int4*>(nidx + i);
  int4 e4 = *reinterpret_cast<const int4*>(eidx + i);
  atomicAdd(&degD[n4.x], 1.0f);
  atomicAdd(&degD[n4.y], 1.0f);
  atomicAdd(&degD[n4.z], 1.0f);
  atomicAdd(&degD[n4.w], 1.0f);
  atomicAdd(&degB[e4.x], 1.0f);
  atomicAdd(&degB[e4.y], 1.0f);
  atomicAdd(&degB[e4.z], 1.0f);
  atomicAdd(&degB[e4.w], 1.0f);
}

// Replace degrees by their safe inverses (in place).
__global__ void hg_invert(float* __restrict__ degD, float* __restrict__ degB) {
  int i = blockIdx.x * blockDim.x + threadIdx.x;
  if (i < kNodes) { float d = degD[i]; degD[i] = d > 0.0f ? 1.0f / d : 0.0f; }
  if (i < kEdges) { float b = degB[i]; degB[i] = b > 0.0f ? 1.0f / b : 0.0f; }
}

// accE[e] += xn[n]  over all nnz  (H^T x)
__global__ void hg_scatter_edges(const int* __restrict__ nidx, const int* __restrict__ eidx,
                                 const float* __restrict__ xn, float* __restrict__ accE) {
  int i = (blockIdx.x * blockDim.x + threadIdx.x) * 4;
  if (i >= kNNZ) return;
  int4 n4 = *reinterpret_cast<const int4*>(nidx + i);
  int4 e4 = *reinterpret_cast<const int4*>(eidx + i);
  atomicAdd(&accE[e4.x], xn[n4.x]);
  atomicAdd(&accE[e4.y], xn[n4.y]);
  atomicAdd(&accE[e4.z], xn[n4.z]);
  atomicAdd(&accE[e4.w], xn[n4.w]);
}

// accN[n] += accE[e] * Binv[e]  over all nnz  (H B^-1 (.))
__global__ void hg_scatter_nodes(const int* __restrict__ nidx, const int* __restrict__ eidx,
                                 const float* __restrict__ accE, const float* __restrict__ Binv,
                                 float* __restrict__ accN) {
  int i = (blockIdx.x * blockDim.x + threadIdx.x) * 4;
  if (i >= kNNZ) return;
  int4 n4 = *reinterpret_cast<const int4*>(nidx + i);
  int4 e4 = *reinterpret_cast<const int4*>(eidx + i);
  atomicAdd(&accN[n4.x], accE[e4.x] * Binv[e4.x]);
  atomicAdd(&accN[n4.y], accE[e4.y] * Binv[e4.y]);
  atomicAdd(&accN[n4.z], accE[e4.z] * Binv[e4.z]);
  atomicAdd(&accN[n4.w], accE[e4.w] * Binv[e4.w]);
}

// ---------------- dense per-node MLP via V_WMMA_F32_16X16X4_F32 -------------
// y[n] = sum_f relu(W1[f]*t1[n] + b1[f]) * W2[f],  t1[n] = accN1[n]*Dinv[n]
// One wave32 per 16-node tile. A = 16x4 relu'd hidden chunk (M=node, K=feat),
// B = 4x16 with W2[k] broadcast across all 16 columns -> every D column holds y.
__global__ void hg_mlp_wmma(const float* __restrict__ accN1, const float* __restrict__ Dinv,
                            const float* __restrict__ W1, const float* __restrict__ b1,
                            const float* __restrict__ W2, float* __restrict__ y) {
  const int lane = threadIdx.x & 31;
  const int wave = threadIdx.x >> 5;
  const int tile = blockIdx.x * (blockDim.x >> 5) + wave;
  const int ntiles = kNodes / 16;          // 6250, exact
  if (tile >= ntiles) return;              // uniform per wave -> EXEC all-1s below
  const int base = tile * 16;

  const int m = base + (lane & 15);        // A/B rows: lanes 0-15 and 16-31 both M=0..15
  const float t = accN1[m] * Dinv[m];      // t1 for this node
  const int k0 = (lane >> 4) << 1;         // K offset within a 4-chunk: 0 (lo half) / 2 (hi half)

  v8f c = {};
#pragma unroll
  for (int ch = 0; ch < kF / 4; ++ch) {    // 32 WMMAs, K = 128
    const int kk = ch * 4 + k0;
    v2f a, b;
    // A[m, kk] / A[m, kk+1]: VGPR0 = K=k0, VGPR1 = K=k0+1 (ISA 32-bit 16x4 layout)
    a.x = fmaxf(fmaf(W1[kk],     t, b1[kk]),     0.0f);
    a.y = fmaxf(fmaf(W1[kk + 1], t, b1[kk + 1]), 0.0f);
    // B[k, n] = W2[kk + k'] for every column n
    b.x = W2[kk];
    b.y = W2[kk + 1];
    c = __builtin_amdgcn_wmma_f32_16x16x4_f32(
        /*neg_a=*/false, a, /*neg_b=*/false, b,
        /*c_mod=*/(short)0, c, /*reuse_a=*/false, /*reuse_b=*/false);
  }

  // D layout: lane l, VGPR v -> (M = v + 8*(l>=16), N = l&15); all N identical.
  if ((lane & 15) == 0) {
    const int mb = base + ((lane >> 4) << 3);
#pragma unroll
    for (int v = 0; v < 8; ++v) y[mb + v] = c[v];
  }
}

__global__ void hg_final(const float* __restrict__ accN2, const float* __restrict__ Dinv,
                         const float* __restrict__ b2, float* __restrict__ out) {
  int i = blockIdx.x * blockDim.x + threadIdx.x;
  if (i < kNodes) out[i] = accN2[i] * Dinv[i] + b2[0];
}

// ---------------------------------------------------------------------------

extern "C" void kernel_launch(void* const* d_in, const int* in_sizes, int n_in,
                              void* d_out, int out_size, void* d_ws, size_t ws_size,
                              hipStream_t stream) {
  (void)in_sizes; (void)n_in; (void)out_size; (void)ws_size;
  const float* x    = (const float*)d_in[0];
  const float* W1   = (const float*)d_in[1];
  const float* b1   = (const float*)d_in[2];
  const float* W2   = (const float*)d_in[3];
  const float* b2   = (const float*)d_in[4];
  const int*   nidx = (const int*)d_in[5];
  const int*   eidx = (const int*)d_in[6];
  float* out = (float*)d_out;

  // Workspace layout (floats): 430000 * 4B = 1.72 MB, fully L2-resident.
  float* ws    = (float*)d_ws;
  float* degD  = ws;            // [100000] degree -> D^-1 after hg_invert
  float* accN1 = ws + 100000;   // [100000] conv1 node accumulator (pre D^-1)
  float* yv    = ws + 200000;   // [100000] y = (relu hidden) @ W2
  float* accN2 = ws + 300000;   // [100000] conv2 node accumulator
  float* degB  = ws + 400000;   // [10000]  degree -> B^-1 after hg_invert
  float* accE1 = ws + 410000;   // [10000]  conv1 edge accumulator
  float* accE2 = ws + 420000;   // [10000]  conv2 edge accumulator
  const int totalWs = 430000;

  const int B = 256;
  const int nq = kNNZ / 4;      // 400000 quad-threads per NNZ pass

  hg_zero<<<(totalWs + B - 1) / B, B, 0, stream>>>(ws, totalWs);
  hg_degree<<<(nq + B - 1) / B, B, 0, stream>>>(nidx, eidx, degD, degB);
  hg_invert<<<(kNodes + B - 1) / B, B, 0, stream>>>(degD, degB);

  // conv1 (scalar propagation of x)
  hg_scatter_edges<<<(nq + B - 1) / B, B, 0, stream>>>(nidx, eidx, x, accE1);
  hg_scatter_nodes<<<(nq + B - 1) / B, B, 0, stream>>>(nidx, eidx, accE1, degB, accN1);

  // per-node 128-wide MLP + projection to scalar y, on the matrix pipe
  const int ntiles = kNodes / 16;      // 6250
  const int wpb = B / 32;              // 8 waves (tiles) per block
  hg_mlp_wmma<<<(ntiles + wpb - 1) / wpb, B, 0, stream>>>(accN1, degD, W1, b1, W2, yv);

  // conv2 (scalar propagation of y)
  hg_scatter_edges<<<(nq + B - 1) / B, B, 0, stream>>>(nidx, eidx, yv, accE2);
  hg_scatter_nodes<<<(nq + B - 1) / B, B, 0, stream>>>(nidx, eidx, accE2, degB, accN2);

  hg_final<<<(kNodes + B - 1) / B, B, 0, stream>>>(accN2, degD, b2, out);
}